// GL_TAGConv_3l_512h_w_k3_gnorm_70111046139980
// MI455X (gfx1250) — compile-verified
//
#include <hip/hip_runtime.h>

typedef __attribute__((ext_vector_type(2))) float v2f;
typedef __attribute__((ext_vector_type(8))) float v8f;

static inline int cdiv_i(int a, int b) { return (a + b - 1) / b; }

// ---------------- edge norm (gcn_norm) ----------------
__global__ __launch_bounds__(256) void k_deg_acc(const int* __restrict__ dst,
                                                 const float* __restrict__ w,
                                                 float* __restrict__ deg, int E) {
  int e = blockIdx.x * 256 + threadIdx.x;
  if (e < E) atomicAdd(&deg[dst[e]], w[e]);
}

__global__ __launch_bounds__(256) void k_dinv(float* __restrict__ deg, int N) {
  int i = blockIdx.x * 256 + threadIdx.x;
  if (i < N) { float d = deg[i]; deg[i] = d > 0.f ? rsqrtf(d) : 0.f; }
}

__global__ __launch_bounds__(256) void k_norm(const int* __restrict__ src, const int* __restrict__ dst,
                                              const float* __restrict__ w, const float* __restrict__ dinv,
                                              float* __restrict__ nrm, int E) {
  int e = blockIdx.x * 256 + threadIdx.x;
  if (e < E) nrm[e] = dinv[src[e]] * w[e] * dinv[dst[e]];
}

// copy x[N,4] into cat1[:, 0:4] (pitch 16)
__global__ __launch_bounds__(256) void k_copy_x(const float* __restrict__ x, float* __restrict__ cat1, int N4) {
  int i = blockIdx.x * 256 + threadIdx.x;
  if (i < N4) cat1[(size_t)(i >> 2) * 16 + (i & 3)] = x[i];
}

// ---------------- 4-wide gather-scale-scatter ----------------
__global__ __launch_bounds__(256) void k_prop4(const float* __restrict__ in, int inPitch, int inOff,
                                               float* __restrict__ out, int outPitch, int outOff,
                                               const int* __restrict__ src, const int* __restrict__ dst,
                                               const float* __restrict__ nrm, int E) {
  int e = blockIdx.x * 256 + threadIdx.x;
  if (e >= E) return;
  int s = src[e], d = dst[e];
  float wv = nrm[e];
  const float* ip = in + (size_t)s * inPitch + inOff;
  float* op = out + (size_t)d * outPitch + outOff;
#pragma unroll
  for (int j = 0; j < 4; ++j) atomicAdd(op + j, wv * ip[j]);
}

// ---------------- 512-wide gather-scale-scatter (L2-resident atomics) ----------------
// block = (128,2): 128 float4 lanes cover 512 channels, 2 edges per block
__global__ __launch_bounds__(256) void k_prop512(const float4* __restrict__ hin, float* __restrict__ hout,
                                                 const int* __restrict__ src, const int* __restrict__ dst,
                                                 const float* __restrict__ nrm, int E) {
  int g = threadIdx.x;                         // float4 group 0..127
  int e = blockIdx.x * 2 + threadIdx.y;
  if (e >= E) return;
  int s = src[e], d = dst[e];
  float wv = nrm[e];
  float4 v = hin[(size_t)s * 128 + g];
  float* o = hout + (size_t)d * 512 + g * 4;
  atomicAdd(o + 0, v.x * wv);
  atomicAdd(o + 1, v.y * wv);
  atomicAdd(o + 2, v.z * wv);
  atomicAdd(o + 3, v.w * wv);
}

// ---------------- WMMA fp32 GEMM: out[M,512] = sum_seg A_seg[M,Kseg] @ B_seg[Kseg,512] ----------------
// Block = 256 threads (8 waves). Block computes a 16-row x 512-col slab.
// A tile staged in LDS once per segment (each A element read from HBM exactly once).
// Wave w owns cols [64w, 64w+64) as 4 accumulator tiles of 16x16, K-step 4 via v_wmma_f32_16x16x4_f32.
__global__ __launch_bounds__(256) void k_gemm_wmma(const float* __restrict__ A0, const float* __restrict__ A1,
                                                   const float* __restrict__ A2, const float* __restrict__ A3,
                                                   int nseg, int Kseg,
                                                   const float* __restrict__ B,      // stacked [nseg*Kseg, 512]
                                                   const float* __restrict__ bias,   // [512]
                                                   float* __restrict__ out, int M, int fuse_elu) {
  __shared__ float lds_a[16 * 512];   // 32 KB (max Kseg = 512)
  const int wave = threadIdx.x >> 5;
  const int lane = threadIdx.x & 31;
  const int m = lane & 15;            // row within 16x16 tile (A), col within tile (B/C)
  const int half = lane >> 4;         // selects K pair / row-half per ISA layout
  const int row0 = blockIdx.x * 16;
  const int col0 = wave * 64;

  v8f acc[4] = {v8f{0.f}, v8f{0.f}, v8f{0.f}, v8f{0.f}};
  const float* As[4] = {A0, A1, A2, A3};

  for (int seg = 0; seg < nseg; ++seg) {
    __syncthreads();
    const float* A = As[seg];
    for (int idx = threadIdx.x; idx < 16 * Kseg; idx += 256) {
      int r = idx / Kseg, c = idx - r * Kseg;
      int row = row0 + r;
      lds_a[idx] = (row < M) ? A[(size_t)row * Kseg + c] : 0.f;
    }
    __syncthreads();

    const float* Bs = B + (size_t)seg * Kseg * 512;
    for (int k = 0; k < Kseg; k += 4) {
      int kk = k + 2 * half;
      v2f a = *(const v2f*)&lds_a[m * Kseg + kk];   // A[m][kk], A[m][kk+1]
#pragma unroll
      for (int t = 0; t < 4; ++t) {
        int col = col0 + 16 * t + m;
        v2f b;
        b.x = Bs[(size_t)kk * 512 + col];
        b.y = Bs[(size_t)(kk + 1) * 512 + col];
        acc[t] = __builtin_amdgcn_wmma_f32_16x16x4_f32(
            /*neg_a=*/false, a, /*neg_b=*/false, b,
            /*c_mod=*/(short)0, acc[t], /*reuse_a=*/false, /*reuse_b=*/false);
      }
    }
  }

  // epilogue: C/D layout — VGPR r: lanes0-15 -> M=r, lanes16-31 -> M=r+8; N = lane&15
#pragma unroll
  for (int t = 0; t < 4; ++t) {
    int col = col0 + 16 * t + m;
    float bv = bias ? bias[col] : 0.f;
#pragma unroll
    for (int r = 0; r < 8; ++r) {
      int row = row0 + r + 8 * half;
      if (row < M) {
        float v = acc[t][r] + bv;
        if (fuse_elu) v = v > 0.f ? v : expm1f(v);
        out[(size_t)row * 512 + col] = v;
      }
    }
  }
}

// ---------------- GraphNorm: fused column sum + sumsq ----------------
__global__ __launch_bounds__(512) void k_colstats(const float* __restrict__ e, int M, float* __restrict__ st) {
  int c = threadIdx.x;  // 512 channels
  float s = 0.f, s2 = 0.f;
  for (int r = blockIdx.x; r < M; r += gridDim.x) {
    float v = e[(size_t)r * 512 + c];
    s += v; s2 += v * v;
  }
  atomicAdd(&st[c], s);
  atomicAdd(&st[512 + c], s2);
}

__global__ __launch_bounds__(512) void k_gn_finalize(const float* __restrict__ st,
                                                     const float* __restrict__ gw, const float* __restrict__ gb,
                                                     const float* __restrict__ ga,
                                                     float* __restrict__ scale, float* __restrict__ shift,
                                                     float invN) {
  int c = threadIdx.x;
  float mean = st[c] * invN;
  float ex2  = st[512 + c] * invN;
  float am   = ga[c] * mean;
  float var  = ex2 - 2.f * am * mean + am * am;   // E[(x - a*m)^2]
  float s    = gw[c] * rsqrtf(var + 1e-5f);
  scale[c] = s;
  shift[c] = gb[c] - s * am;
}

__global__ __launch_bounds__(256) void k_gn_apply(const float* __restrict__ e, const float* __restrict__ sc,
                                                  const float* __restrict__ sh, float* __restrict__ h, int total) {
  int i = blockIdx.x * 256 + threadIdx.x;
  if (i < total) { int c = i & 511; h[i] = fmaf(e[i], sc[c], sh[c]); }
}

// ---------------- layer 3 projection: z[N,16] = g[N,512] @ W3[k][512][4] ----------------
__global__ __launch_bounds__(256) void k_proj16(const float* __restrict__ g, const float* __restrict__ W3,
                                                float* __restrict__ z, int M) {
  __shared__ float gs[16 * 512];   // 32 KB
  __shared__ float ws[4 * 512 * 4]; // 32 KB (all of W3)
  int row0 = blockIdx.x * 16;
  for (int idx = threadIdx.x; idx < 8192; idx += 256) {
    int r = idx >> 9, c = idx & 511;
    int row = row0 + r;
    gs[idx] = (row < M) ? g[(size_t)row * 512 + c] : 0.f;
    ws[idx] = W3[idx];
  }
  __syncthreads();
  int t = threadIdx.x >> 4;   // node 0..15
  int o = threadIdx.x & 15;   // output 0..15 (o = 4k + j)
  int k = o >> 2, j = o & 3;
  const float* gp = gs + t * 512;
  const float* wp = ws + k * 2048 + j;  // W3[k][i][j] at k*2048 + i*4 + j
  float acc = 0.f;
  for (int i = 0; i < 512; ++i) acc = fmaf(gp[i], wp[i * 4], acc);
  int row = row0 + t;
  if (row < M) z[(size_t)row * 16 + o] = acc;
}

// t[N,4] += z[:, off:off+4]
__global__ __launch_bounds__(256) void k_addz(float* __restrict__ t, const float* __restrict__ z, int off, int N4) {
  int i = blockIdx.x * 256 + threadIdx.x;
  if (i < N4) t[i] += z[(size_t)(i >> 2) * 16 + off + (i & 3)];
}

// out[N,4] += z[:,0:4] + b3
__global__ __launch_bounds__(256) void k_final(float* __restrict__ out, const float* __restrict__ z,
                                               const float* __restrict__ b3, int N4) {
  int i = blockIdx.x * 256 + threadIdx.x;
  if (i < N4) out[i] += z[(size_t)(i >> 2) * 16 + (i & 3)] + b3[i & 3];
}

// =====================================================================

extern "C" void kernel_launch(void* const* d_in, const int* in_sizes, int n_in,
                              void* d_out, int out_size, void* d_ws, size_t ws_size,
                              hipStream_t stream) {
  const float* x    = (const float*)d_in[0];
  const int*   ei   = (const int*)d_in[1];
  const float* ew   = (const float*)d_in[2];
  const float* W1   = (const float*)d_in[3];   // [4,4,512]  == stacked [16,512]
  const float* b1   = (const float*)d_in[4];
  const float* W2   = (const float*)d_in[5];   // [4,512,512] == stacked [2048,512]
  const float* b2   = (const float*)d_in[6];
  const float* W3   = (const float*)d_in[7];   // [4,512,4]
  const float* b3   = (const float*)d_in[8];
  const float* gn1w = (const float*)d_in[9];
  const float* gn1b = (const float*)d_in[10];
  const float* gn1a = (const float*)d_in[11];
  const float* gn2w = (const float*)d_in[12];
  const float* gn2b = (const float*)d_in[13];
  const float* gn2a = (const float*)d_in[14];

  const int N = in_sizes[0] / 4;
  const int E = in_sizes[2];
  const int* src = ei;
  const int* dst = ei + E;
  float* outp = (float*)d_out;

  // ---- workspace layout (floats, 64-float aligned slabs) ----
  float* ws = (float*)d_ws;
  size_t o = 0;
  auto take = [&](size_t nfloats) { float* p = ws + o; o += (nfloats + 63) & ~(size_t)63; return p; };
  float* deg   = take(N);
  float* nrm   = take(E);
  float* cat1  = take((size_t)N * 16);
  float* z     = take((size_t)N * 16);
  float* t4a   = take((size_t)N * 4);
  float* t4b   = take((size_t)N * 4);
  float* stats = take(1024);
  float* scale = take(512);
  float* shift = take(512);
  float* ebuf  = take((size_t)N * 512);
  float* h     = take((size_t)N * 512);
  float* h1    = take((size_t)N * 512);
  float* h2    = take((size_t)N * 512);
  float* h3    = take((size_t)N * 512);
  (void)ws_size; (void)n_in; (void)out_size;

  const int gE = cdiv_i(E, 256);
  const int gN = cdiv_i(N, 256);
  const int gN4 = cdiv_i(N * 4, 256);
  const int gRows = cdiv_i(N, 16);
  const dim3 pb(128, 2, 1);
  const int gProp = cdiv_i(E, 2);
  const size_t fbytes512 = (size_t)N * 512 * sizeof(float);

  // ---- gcn_norm ----
  hipMemsetAsync(deg, 0, (size_t)N * sizeof(float), stream);
  k_deg_acc<<<gE, 256, 0, stream>>>(dst, ew, deg, E);
  k_dinv<<<gN, 256, 0, stream>>>(deg, N);
  k_norm<<<gE, 256, 0, stream>>>(src, dst, ew, deg, nrm, E);

  // ---- layer 1: propagate x at 4-dim into cat1 = [x, Ax, A2x, A3x] ----
  hipMemsetAsync(cat1, 0, (size_t)N * 16 * sizeof(float), stream);
  k_copy_x<<<gN4, 256, 0, stream>>>(x, cat1, N * 4);
  k_prop4<<<gE, 256, 0, stream>>>(cat1, 16, 0, cat1, 16, 4,  src, dst, nrm, E);
  k_prop4<<<gE, 256, 0, stream>>>(cat1, 16, 4, cat1, 16, 8,  src, dst, nrm, E);
  k_prop4<<<gE, 256, 0, stream>>>(cat1, 16, 8, cat1, 16, 12, src, dst, nrm, E);
  // GEMM [N,16]@[16,512] + b1, fused ELU
  k_gemm_wmma<<<gRows, 256, 0, stream>>>(cat1, nullptr, nullptr, nullptr, 1, 16, W1, b1, ebuf, N, 1);
  // GraphNorm 1 -> h
  hipMemsetAsync(stats, 0, 1024 * sizeof(float), stream);
  k_colstats<<<256, 512, 0, stream>>>(ebuf, N, stats);
  k_gn_finalize<<<1, 512, 0, stream>>>(stats, gn1w, gn1b, gn1a, scale, shift, 1.f / (float)N);
  k_gn_apply<<<cdiv_i(N * 512, 256), 256, 0, stream>>>(ebuf, scale, shift, h, N * 512);

  // ---- layer 2: three 512-dim hops (L2-resident atomics) ----
  hipMemsetAsync(h1, 0, fbytes512, stream);
  k_prop512<<<gProp, pb, 0, stream>>>((const float4*)h,  h1, src, dst, nrm, E);
  hipMemsetAsync(h2, 0, fbytes512, stream);
  k_prop512<<<gProp, pb, 0, stream>>>((const float4*)h1, h2, src, dst, nrm, E);
  hipMemsetAsync(h3, 0, fbytes512, stream);
  k_prop512<<<gProp, pb, 0, stream>>>((const float4*)h2, h3, src, dst, nrm, E);
  // GEMM sum_k h_k @ W2[k] + b2, fused ELU  (K = 4 x 512, fp32 WMMA)
  k_gemm_wmma<<<gRows, 256, 0, stream>>>(h, h1, h2, h3, 4, 512, W2, b2, ebuf, N, 1);
  // GraphNorm 2 -> h (g)
  hipMemsetAsync(stats, 0, 1024 * sizeof(float), stream);
  k_colstats<<<256, 512, 0, stream>>>(ebuf, N, stats);
  k_gn_finalize<<<1, 512, 0, stream>>>(stats, gn2w, gn2b, gn2a, scale, shift, 1.f / (float)N);
  k_gn_apply<<<cdiv_i(N * 512, 256), 256, 0, stream>>>(ebuf, scale, shift, h, N * 512);

  // ---- layer 3: project first (z_k = g @ W3[k]), then Horner 4-dim propagation ----
  k_proj16<<<gRows, 256, 0, stream>>>(h, W3, z, N);
  hipMemsetAsync(t4a, 0, (size_t)N * 4 * sizeof(float), stream);
  k_prop4<<<gE, 256, 0, stream>>>(z, 16, 12, t4a, 4, 0, src, dst, nrm, E);   // A z3
  k_addz<<<gN4, 256, 0, stream>>>(t4a, z, 8, N * 4);                         // + z2
  hipMemsetAsync(t4b, 0, (size_t)N * 4 * sizeof(float), stream);
  k_prop4<<<gE, 256, 0, stream>>>(t4a, 4, 0, t4b, 4, 0, src, dst, nrm, E);   // A(...)
  k_addz<<<gN4, 256, 0, stream>>>(t4b, z, 4, N * 4);                         // + z1
  hipMemsetAsync(outp, 0, (size_t)N * 4 * sizeof(float), stream);
  k_prop4<<<gE, 256, 0, stream>>>(t4b, 4, 0, outp, 4, 0, src, dst, nrm, E);  // A(...)
  k_final<<<gN4, 256, 0, stream>>>(outp, z, b3, N * 4);                      // + z0 + b3
}